// SSDLayer_85126251807528
// MI455X (gfx1250) — compile-verified
//
#include <hip/hip_runtime.h>
#include <stdint.h>

// ---------------------------------------------------------------------------
// SSD MultiBox loss for MI455X (gfx1250).
//  Pass 1: stream p/loc_t/conf_t once (BW-bound, ~159MB @ 23.3TB/s ≈ 7us),
//          emit loss_c[] (8MB, L2-resident), accumulate num_pos / loc / pos-CE.
//  Pass 2: per row (64 blocks): TDM the 128KB row into LDS (320KB/WGP),
//          32-step radix select of the K-th largest loss_c, exact top-K sum
//          via  sum_{v>t} v + (K - cnt_gt)*t   (ties share the same value).
//          Block reductions use a 5x v_wmma_f32_16x16x4_f32 wave32 all-reduce.
//  Pass 3: scalar finalize.
// ---------------------------------------------------------------------------

#define BDIM   64
#define PDIM   32768
#define NATTR  14
#define NCLS   10
#define NEGPOS 3

typedef uint32_t u32;
typedef uint64_t u64;
typedef __attribute__((ext_vector_type(4))) u32 u32x4;
typedef __attribute__((ext_vector_type(8))) u32 u32x8;
typedef __attribute__((ext_vector_type(2))) float v2f;
typedef __attribute__((ext_vector_type(8))) float v8f;

// ws layout: [0]=loss_loc (f32), [1]=pos_ce (f32), [2]=neg_sum (f32),
//            [3]=num_pos (i32), loss_c[] at byte offset 1024.

__global__ void ssd_init(float* acc) {
    acc[0] = 0.0f; acc[1] = 0.0f; acc[2] = 0.0f;
    ((int*)acc)[3] = 0;
}

__device__ __forceinline__ float smooth_l1(float d) {
    float a = fabsf(d);
    return (a < 1.0f) ? 0.5f * d * d : a - 0.5f;
}

// Exact wave32 all-reduce (sum) in 5 V_WMMA_F32_16X16X4_F32 ops.
// Step 1: A holds lane partials at K0 (lanes 0-15 -> rows, K=0) and K2
// (lanes 16-31), zeros at K1/K3; B = ones  =>  D1[m][n] = p_m + p_{m+16}.
// Steps 2-5: feed D1 VGPR pairs back as A against ones, accumulating; the
// four row-sums cover each s_m exactly once  =>  every lane gets the total.
// Must be called with EXEC all ones (convergent, full block active).
__device__ __forceinline__ float wave_reduce_sum(float x) {
    v2f ones; ones.x = 1.0f; ones.y = 1.0f;
    v2f a;    a.x = x;       a.y = 0.0f;
    v8f z = {};
    v8f d1 = __builtin_amdgcn_wmma_f32_16x16x4_f32(
        false, a, false, ones, (short)0, z, false, false);
    v8f acc = {};
    #pragma unroll
    for (int i = 0; i < 4; ++i) {
        v2f ai; ai.x = d1[2 * i]; ai.y = d1[2 * i + 1];
        acc = __builtin_amdgcn_wmma_f32_16x16x4_f32(
            false, ai, false, ones, (short)0, acc, false, false);
    }
    return acc[0];
}

__global__ __launch_bounds__(256)
void ssd_pass1(const float* __restrict__ p,
               const float* __restrict__ loc_t,
               const int*   __restrict__ conf_t,
               float*       __restrict__ loss_c,
               float*       __restrict__ acc) {
    __shared__ float s_loc, s_ce, s_cnt;
    if (threadIdx.x == 0) { s_loc = 0.0f; s_ce = 0.0f; s_cnt = 0.0f; }
    __syncthreads();

    const int N      = BDIM * PDIM;
    const int stride = gridDim.x * blockDim.x;
    float t_loc = 0.0f, t_ce = 0.0f, t_cnt = 0.0f;

    for (int idx = blockIdx.x * blockDim.x + threadIdx.x; idx < N; idx += stride) {
        // Prefetch next grid-stride tile (speculative, OOB-safe) -> global_prefetch_b8
        __builtin_prefetch(p + (u64)(idx + stride) * NATTR, 0, 1);

        // 14 floats per anchor, 8B-aligned float2 loads (56B stride).
        const float2* e = (const float2*)(p + (u64)idx * NATTR);
        float2 d0 = e[0], d1 = e[1];
        float2 c0 = e[2], c1 = e[3], c2 = e[4], c3 = e[5], c4 = e[6];
        const float4 lt = ((const float4*)loc_t)[idx];
        const int ct = conf_t[idx];
        const bool pos = ct > 0;

        float sl = smooth_l1(d0.x - lt.x) + smooth_l1(d0.y - lt.y) +
                   smooth_l1(d1.x - lt.z) + smooth_l1(d1.y - lt.w);

        float cls[NCLS] = {c0.x, c0.y, c1.x, c1.y, c2.x,
                           c2.y, c3.x, c3.y, c4.x, c4.y};
        float m = cls[0];
        #pragma unroll
        for (int i = 1; i < NCLS; ++i) m = fmaxf(m, cls[i]);
        float se = 0.0f;
        #pragma unroll
        for (int i = 0; i < NCLS; ++i) se += __expf(cls[i] - m);
        const float lse = m + __logf(se);

        float tgt = cls[0];
        #pragma unroll
        for (int c = 1; c < NCLS; ++c) tgt = (ct == c) ? cls[c] : tgt;
        const float ce = lse - tgt;

        loss_c[idx] = pos ? 0.0f : ce;          // ce >= 0, so uint order == float order
        if (pos) { t_cnt += 1.0f; t_loc += sl; t_ce += ce; }
    }

    // Wave32 WMMA all-reduce, then one ds_add_f32 per wave.
    const float w_loc = wave_reduce_sum(t_loc);
    const float w_ce  = wave_reduce_sum(t_ce);
    const float w_cnt = wave_reduce_sum(t_cnt);
    if ((threadIdx.x & 31) == 0) {
        atomicAdd(&s_loc, w_loc);
        atomicAdd(&s_ce,  w_ce);
        atomicAdd(&s_cnt, w_cnt);
    }
    __syncthreads();
    if (threadIdx.x == 0) {
        atomicAdd(&acc[0], s_loc);
        atomicAdd(&acc[1], s_ce);
        atomicAdd((int*)acc + 3, (int)s_cnt);
    }
}

// One block per batch row. Dynamic LDS: PDIM u32 of row data + 16 u32 scratch.
__global__ __launch_bounds__(1024)
void ssd_pass2(const float* __restrict__ loss_c, float* __restrict__ acc) {
    extern __shared__ u32 smem[];          // no static LDS => dynamic base at LDS addr 0
    const int tid = threadIdx.x;
    const int b   = blockIdx.x;

    const int np = ((const int*)acc)[3];
    long long K = (long long)NEGPOS * (long long)np;
    const long long cap = (long long)PDIM - (long long)np;
    if (cap < K) K = cap;
    if (K <= 0) return;                    // uniform exit: no negatives mined
    if (K > PDIM) K = PDIM;

    // --- Tensor Data Mover: copy 32768 f32 (128KB) row to LDS address 0 ------------
    if (tid == 0) {
        const u64 ga = (u64)(const void*)(loss_c + (u64)b * PDIM);
        u32x4 g0;
        g0[0] = 1u;                                        // count=1, user, !gather
        g0[1] = 0u;                                        // LDS destination address is 0
        g0[2] = (u32)ga;                                   // global_addr[31:0]
        g0[3] = (u32)((ga >> 32) & 0x01FFFFFFu)            // global_addr[56:32]
              | (2u << 30);                                // type = 2 ("image")
        u32x8 g1;
        const u32 td0 = PDIM;                              // tensor_dim0
        g1[0] = (2u << 16);                                // wg_mask=0, data_size=4B
        g1[1] = (td0 & 0xFFFFu) << 16;                     // tensor_dim0[15:0]
        g1[2] = ((td0 >> 16) & 0xFFFFu)                    // tensor_dim0[31:16]
              | (1u << 16);                                // tensor_dim1 = 1
        g1[3] = ((td0 & 0xFFFFu) << 16);                   // tile_dim0 = 32768
        g1[4] = 0u;                                        // tile_dim1/2 unused (1-D)
        g1[5] = td0;                                       // tensor_dim0_stride[31:0]
        g1[6] = 0u;                                        // stride hi / dim1_stride lo
        g1[7] = 0u;
        asm volatile("tensor_load_to_lds %0, %1" :: "s"(g0), "s"(g1) : "memory");
        __builtin_amdgcn_s_wait_tensorcnt(0);
    }
    __syncthreads();

    float* s_fcnt = (float*)smem + PDIM;       // shared float counter
    float* s_fsum = (float*)smem + PDIM + 1;   // shared float value sum

    // --- radix select of K-th largest (uint order == float order, v >= 0) ---
    u32 thr = 0;
    for (int bit = 31; bit >= 0; --bit) {
        const u32 cand = thr | (1u << bit);
        __syncthreads();
        if (tid == 0) *s_fcnt = 0.0f;
        __syncthreads();
        float c = 0.0f;
        for (int i = tid; i < PDIM; i += blockDim.x) c += (smem[i] >= cand) ? 1.0f : 0.0f;
        const float wc = wave_reduce_sum(c);   // 5x v_wmma_f32_16x16x4_f32
        if ((tid & 31) == 0) atomicAdd(s_fcnt, wc);
        __syncthreads();
        if ((long long)(*s_fcnt) >= K) thr = cand;   // exact: count <= 32768 < 2^24
    }

    // --- exact top-K sum: sum_{v>thr} v + (K - cnt_gt) * thr ----------------
    __syncthreads();
    if (tid == 0) { *s_fcnt = 0.0f; *s_fsum = 0.0f; }
    __syncthreads();
    float vs = 0.0f, cg = 0.0f;
    for (int i = tid; i < PDIM; i += blockDim.x) {
        const u32 x = smem[i];
        if (x > thr) { vs += __uint_as_float(x); cg += 1.0f; }
    }
    const float wvs = wave_reduce_sum(vs);
    const float wcg = wave_reduce_sum(cg);
    if ((tid & 31) == 0) { atomicAdd(s_fsum, wvs); atomicAdd(s_fcnt, wcg); }
    __syncthreads();
    if (tid == 0) {
        const float total = *s_fsum
                          + (float)(K - (long long)(*s_fcnt)) * __uint_as_float(thr);
        atomicAdd(&acc[2], total);
    }
}

__global__ void ssd_finalize(const float* __restrict__ acc, float* __restrict__ out) {
    const int np = ((const int*)acc)[3];
    out[0] = (acc[0] + acc[1] + acc[2]) / (float)np;
}

extern "C" void kernel_launch(void* const* d_in, const int* in_sizes, int n_in,
                              void* d_out, int out_size, void* d_ws, size_t ws_size,
                              hipStream_t stream) {
    const float* p      = (const float*)d_in[0];
    const float* loc_t  = (const float*)d_in[1];
    const int*   conf_t = (const int*)d_in[2];
    float* acc    = (float*)d_ws;
    float* loss_c = (float*)((char*)d_ws + 1024);

    ssd_init<<<1, 1, 0, stream>>>(acc);
    ssd_pass1<<<2048, 256, 0, stream>>>(p, loc_t, conf_t, loss_c, acc);
    ssd_pass2<<<BDIM, 1024, (PDIM + 16) * sizeof(u32), stream>>>(loss_c, acc);
    ssd_finalize<<<1, 1, 0, stream>>>(acc, (float*)d_out);
}